// CosSim_Loss_87393994539199
// MI455X (gfx1250) — compile-verified
//
#include <hip/hip_runtime.h>
#include <hip/hip_bf16.h>

// Problem geometry (fixed by the reference): pred[32][8][512*512] fp32
#define L_ELEMS   (512 * 512)   // 262144 elements per map
#define N_MAPS    8
#define N_BATCH   32
#define PAIRS     (N_BATCH / 2) // 16 batch-pairs -> 16 rows per pair, contiguous
#define WG_PER_PAIR 16
#define WAVES_PER_WG 8
#define WAVE_SLAB (L_ELEMS / (WG_PER_PAIR * WAVES_PER_WG)) // 2048 K-elems per wave

typedef __attribute__((ext_vector_type(2))) float v2f;
typedef __attribute__((ext_vector_type(8))) float v8f;

__device__ __forceinline__ v8f wmma_f32_16x16x4(v2f a, v2f b, v8f c) {
#if __has_builtin(__builtin_amdgcn_wmma_f32_16x16x4_f32)
    // 8 args: (neg_a, A, neg_b, B, c_mod, C, reuse_a, reuse_b)
    return __builtin_amdgcn_wmma_f32_16x16x4_f32(false, a, false, b,
                                                 (short)0, c, false, false);
#else
    asm volatile("v_wmma_f32_16x16x4_f32 %0, %1, %2, %0"
                 : "+v"(c) : "v"(a), "v"(b));
    return c;
#endif
}

__global__ void CosSim_zero_ws(float* ws) {
    for (int i = threadIdx.x; i < N_BATCH * N_MAPS * N_MAPS; i += blockDim.x)
        ws[i] = 0.0f;
}

// Phase 1: accumulate S[b][i][j] = <x_bi, x_bj> for all i,j via fp32 WMMA.
// grid = (WG_PER_PAIR, PAIRS), block = 256 (8 wave32s).
// A 16-row "super batch" = rows of batch 2p (M=0..7) and batch 2p+1 (M=8..15),
// which are 16 contiguous rows of pred. D = A * A^T per K-chunk of 4.
__global__ __launch_bounds__(256)
void CosSim_gram_wmma(const float* __restrict__ pred, float* __restrict__ ws) {
    __shared__ float red[2 * N_MAPS * N_MAPS]; // 128: [batch 2p | batch 2p+1]

    const int tid  = threadIdx.x;
    const int wave = tid >> 5;
    const int lane = tid & 31;
    const int pair = blockIdx.y;

    if (tid < 128) red[tid] = 0.0f;
    __syncthreads();

    // A-matrix f32 16x4 layout: lanes 0-15 hold M=lane with {K0,K1} in v0/v1,
    // lanes 16-31 hold M=lane-16 with {K2,K3}.  B mirrors it with N=lane&15,
    // so for a Gram matrix the A and B register images are identical.
    const int r  = lane & 15;        // row within the 16-row super batch
    const int hi = lane >> 4;        // 0 -> K{0,1}, 1 -> K{2,3}
    const float* rowp = pred + ((size_t)pair * 16 + (size_t)r) * (size_t)L_ELEMS;

    size_t k = ((size_t)blockIdx.x * WAVES_PER_WG + (size_t)wave) * WAVE_SLAB
             + (size_t)(2 * hi);

    v8f c = {0.f, 0.f, 0.f, 0.f, 0.f, 0.f, 0.f, 0.f};

    #pragma unroll 8
    for (int it = 0; it < WAVE_SLAB / 4; ++it) {
        v2f a = *(const v2f*)(rowp + k);   // 8B load, 8B-aligned
        c = wmma_f32_16x16x4(a, a, c);     // C[M][N] += sum_k X[M][k]*X[N][k]
        k += 4;
    }

    // C/D layout: lane t holds N = t&15; VGPR v holds M = v + 8*(t>=16).
    // Batch 2p   Gram = M 0..7,  N 0..7  -> lanes 0..7,  i=v, j=lane
    // Batch 2p+1 Gram = M 8..15, N 8..15 -> lanes 24..31, i=v, j=lane-24
    if (lane < 8) {
        #pragma unroll
        for (int v = 0; v < 8; ++v)
            atomicAdd(&red[v * 8 + lane], c[v]);            // ds_add_f32
    } else if (lane >= 24) {
        #pragma unroll
        for (int v = 0; v < 8; ++v)
            atomicAdd(&red[64 + v * 8 + (lane - 24)], c[v]);
    }
    __syncthreads();

    if (tid < 128)
        atomicAdd(&ws[(size_t)pair * 128 + tid], red[tid]); // global_atomic_add_f32
}

// Phase 2: one wave; lane b handles batch b.
// result = ( sum_b sum_{i!=j} S_ij / (max(sqrt(S_ii),eps)*max(sqrt(S_jj),eps))
//            + 8 per batch (unit diagonal) ) / 2048
__global__ void CosSim_finalize(const float* __restrict__ ws, float* __restrict__ out) {
    const int lane = threadIdx.x;   // 0..31 == batch index
    const float* S = ws + (size_t)lane * 64;

    float d[8];
    #pragma unroll
    for (int i = 0; i < 8; ++i)
        d[i] = fmaxf(sqrtf(S[i * 8 + i]), 1e-8f);

    float acc = 8.0f;               // diagonal forced to exactly 1.0
    #pragma unroll
    for (int i = 0; i < 8; ++i) {
        #pragma unroll
        for (int j = 0; j < 8; ++j) {
            if (i != j) acc += S[i * 8 + j] / (d[i] * d[j]);
        }
    }

    for (int o = 16; o > 0; o >>= 1)
        acc += __shfl_down(acc, o, 32);

    if (lane == 0)
        out[0] = acc / (float)(N_BATCH * N_MAPS * N_MAPS);
}

extern "C" void kernel_launch(void* const* d_in, const int* in_sizes, int n_in,
                              void* d_out, int out_size, void* d_ws, size_t ws_size,
                              hipStream_t stream) {
    const float* pred = (const float*)d_in[0];
    float* out = (float*)d_out;
    float* ws  = (float*)d_ws;   // needs 32*8*8*4 = 8 KB

    CosSim_zero_ws<<<1, 256, 0, stream>>>(ws);
    CosSim_gram_wmma<<<dim3(WG_PER_PAIR, PAIRS), 256, 0, stream>>>(pred, ws);
    CosSim_finalize<<<1, 32, 0, stream>>>(ws, out);
}